// MultiInnerProductDecoder4_15367392985219
// MI455X (gfx1250) — compile-verified
//
#include <hip/hip_runtime.h>
#include <hip/hip_bf16.h>
#include <math.h>

// Problem constants (match reference)
#define NUM_ET   8
#define N_NODES  100000
#define IN_DIM   128
#define N_EDGES  500000

typedef __attribute__((ext_vector_type(2))) float v2f;
typedef __attribute__((ext_vector_type(4))) float v4f;
typedef __attribute__((ext_vector_type(8))) float v8f;

// One wave (32 lanes) handles 16 edges of one edge-type via the fp32 WMMA
// diagonal trick (V_WMMA_F32_16X16X4_F32, full fp32 precision).
//
// Per "group" g (0..15) each lane loads a float4 of its edge's row at
// dim = g*8 + (lane>>4)*4, i.e. half 0 owns dims [8g..8g+3], half 1 owns
// [8g+4..8g+7]. Dot products are order-invariant, so group g feeds TWO
// WMMA K-steps:
//   step0: K0,K1 = dims 8g,8g+1 (half0) ; K2,K3 = dims 8g+4,8g+5 (half1)
//   step1: K0,K1 = dims 8g+2,8g+3       ; K2,K3 = dims 8g+6,8g+7
// A and B pair identical dims at identical K slots -> diag(C) is exact.
// 16 b128 gathers per operand instead of 32 b64 -> half the memory requests.
__global__ __launch_bounds__(256) void mipd_wmma_kernel(
    const float* __restrict__ z,              // [T, N, D] fp32
    const float* __restrict__ weight,         // [T, D]    fp32
    const long long* __restrict__ edge_index, // [T, 2, E] int64
    float* __restrict__ out)                  // [2 * T * E] fp32
{
    __shared__ float ws[IN_DIM];

    const int t = blockIdx.y;                       // edge type, uniform per block
    if (threadIdx.x < IN_DIM)
        ws[threadIdx.x] = weight[t * IN_DIM + threadIdx.x];
    __syncthreads();

    const int lane = threadIdx.x & 31;
    const int wave = threadIdx.x >> 5;
    const int groupsPerType = N_EDGES / 16;         // 31250 (exact, no tail)
    const int group = blockIdx.x * 8 + wave;
    if (group >= groupsPerType) return;             // wave-uniform exit, EXEC stays full

    const int r    = lane & 15;                     // edge slot within group (M == N)
    const int half = lane >> 4;                     // which 4-dim chunk of each 8-dim group
    const long long e = (long long)group * 16 + r;

    const size_t eiBase = (size_t)t * 2 * N_EDGES;
    const long long srcN = edge_index[eiBase + (size_t)e];
    const long long dstN = edge_index[eiBase + N_EDGES + (size_t)e];

    const float* __restrict__ zsrc = z + ((size_t)t * N_NODES + (size_t)srcN) * IN_DIM;
    const float* __restrict__ zdst = z + ((size_t)t * N_NODES + (size_t)dstN) * IN_DIM;

    v8f c0 = {};
    v8f c1 = {};
    const int off0 = half * 4;

    #pragma unroll 4
    for (int g = 0; g < IN_DIM / 8; ++g) {          // 16 groups -> 32 WMMA steps
        const int off = g * 8 + off0;               // 16B aligned
        v4f sv = *(const v4f*)(zsrc + off);         // global_load_b128 (coalesced per row)
        v4f dv = *(const v4f*)(zdst + off);         // global_load_b128
        v4f wv = *(const v4f*)(ws + off);           // ds_load_b128
        v2f a0, a1, b0, b1;
        a0.x = sv.x * wv.x;  a0.y = sv.y * wv.y;    // fold w into A -> diag = src*w*dst
        a1.x = sv.z * wv.z;  a1.y = sv.w * wv.w;
        b0.x = dv.x;         b0.y = dv.y;
        b1.x = dv.z;         b1.y = dv.w;
        c0 = __builtin_amdgcn_wmma_f32_16x16x4_f32(
                false, a0, false, b0, (short)0, c0, false, false);
        c1 = __builtin_amdgcn_wmma_f32_16x16x4_f32(
                false, a1, false, b1, (short)0, c1, false, false);
    }
    v8f c = c0 + c1;                                // merge the two XDL chains

    // Diagonal extraction. C layout (16x16 f32): VGPR v, lanes 0-15 -> (M=v, N=lane),
    // lanes 16-31 -> (M=v+8, N=lane-16). Element (d,d): lane d holds it in c[d] (d<8),
    // lane 16+d holds it in c[d-8] (d>=8).
    const int vg = r - 8 * half;                    // valid diag owner iff 0..7
    float dval = c[0];
    dval = (vg == 1) ? c[1] : dval;
    dval = (vg == 2) ? c[2] : dval;
    dval = (vg == 3) ? c[3] : dval;
    dval = (vg == 4) ? c[4] : dval;
    dval = (vg == 5) ? c[5] : dval;
    dval = (vg == 6) ? c[6] : dval;
    dval = (vg == 7) ? c[7] : dval;

    // Route diag(r) to writer lane r (lanes 0..15): source lane is r (r<8) or 16+r (r>=8).
    const int srcLane = (r < 8) ? r : (16 + r);
    const float logit = __shfl(dval, srcLane, 32);

    if (half == 0) {
        const float s = 1.0f / (1.0f + __expf(-logit));
        const size_t idx = (size_t)t * N_EDGES + (size_t)e;
        // Streaming results: non-temporal so we don't evict the hot z slice from L2.
        __builtin_nontemporal_store(s, &out[idx]);                                   // edge_type_values
        __builtin_nontemporal_store(s, &out[(size_t)NUM_ET * N_EDGES + idx]);        // score (same order)
    }
}

extern "C" void kernel_launch(void* const* d_in, const int* in_sizes, int n_in,
                              void* d_out, int out_size, void* d_ws, size_t ws_size,
                              hipStream_t stream) {
    (void)in_sizes; (void)n_in; (void)out_size; (void)d_ws; (void)ws_size;

    const float*     z  = (const float*)d_in[0];             // [8, 100000, 128] fp32
    const float*     w  = (const float*)d_in[1];             // [8, 128] fp32
    const long long* ei = (const long long*)d_in[2];         // [8, 2, 500000] int64
    float*           out = (float*)d_out;                    // [2 * 8 * 500000] fp32

    const int groupsPerType = N_EDGES / 16;                  // 31250
    dim3 grid((groupsPerType + 7) / 8, NUM_ET, 1);           // 3907 x 8 blocks
    dim3 block(256, 1, 1);                                   // 8 waves/block
    mipd_wmma_kernel<<<grid, block, 0, stream>>>(z, w, ei, out);
}